// MMDLoss_34316788695782
// MI455X (gfx1250) — compile-verified
//
#include <hip/hip_runtime.h>
#include <math.h>

// ---------------------------------------------------------------------------
// MMD loss, MI455X (gfx1250, wave32, WMMA).
//
// Roofline: G = X X^T with X = concat(source,target) in R^{8192x256}.
//   FLOPs: 2*8192^2*256 ~ 34 GFLOP (done as 2 triangular passes ~ 1 full GEMM)
//   Bytes: inputs are 8 MB -> fully L2 resident (192 MB); compute bound.
//   => use V_WMMA_F32_16X16X4_F32 (fp32 matrix pipe, full precision).
// ---------------------------------------------------------------------------

typedef __attribute__((ext_vector_type(2))) float v2f;
typedef __attribute__((ext_vector_type(4))) float v4f;
typedef __attribute__((ext_vector_type(8))) float v8f;

#define DDIM   256        // feature dim
#define NHALF  4096       // rows per side
#define NTOT   8192       // total rows
#define BT     128        // block tile (rows == cols)
#define KT     32         // K slice staged in LDS
#define LDSS   (KT + 4)   // padded LDS row stride (36 -> conflict-free frags)
#define NB     (NTOT / BT)          // 64 block-rows
#define NBLK   (NB * (NB + 1) / 2)  // 2080 lower-tri blocks

// ws layout in floats
#define WS_BW   0
#define WS_P1   16
#define WS_P2   4096
#define WS_SQ   8192

// ---------------------------------------------------------------- sq kernel
__global__ __launch_bounds__(256)
void mmd_sq(const float* __restrict__ src, const float* __restrict__ tgt,
            float* __restrict__ sq) {
  int row  = blockIdx.x * 8 + (threadIdx.x >> 5);
  int lane = threadIdx.x & 31;
  const float* p = (row < NHALF) ? src + (size_t)row * DDIM
                                 : tgt + (size_t)(row - NHALF) * DDIM;
  v4f a = *(const v4f*)(p + lane * 8);
  v4f b = *(const v4f*)(p + lane * 8 + 4);
  float s = a[0]*a[0] + a[1]*a[1] + a[2]*a[2] + a[3]*a[3]
          + b[0]*b[0] + b[1]*b[1] + b[2]*b[2] + b[3]*b[3];
#pragma unroll
  for (int off = 16; off > 0; off >>= 1) s += __shfl_xor(s, off, 32);
  if (lane == 0) sq[row] = s;
}

// ------------------------------------------------------- triangular decode
__device__ __forceinline__ void decode_tri(int t, int& bi, int& bj) {
  int i = (int)((sqrtf(8.0f * (float)t + 1.0f) - 1.0f) * 0.5f);
  while ((i + 1) * (i + 2) / 2 <= t) ++i;
  while (i * (i + 1) / 2 > t) --i;
  bi = i;
  bj = t - i * (i + 1) / 2;
}

// ------------------------------------------------------------- GEMM passes
// PASS 1: partial[t] = sum of relu(l2) over block (x2 for off-diagonal)
// PASS 2: partial[t] = sum of 5-Gaussian kernel values over block (unweighted)
template <int PASS>
__global__ __launch_bounds__(256)
void mmd_gemm(const float* __restrict__ src, const float* __restrict__ tgt,
              const float* __restrict__ sq, float* __restrict__ partial,
              const float* __restrict__ bwp) {
  __shared__ float lA[BT * LDSS];
  __shared__ float lB[BT * LDSS];
  __shared__ float sqA[BT];
  __shared__ float sqB[BT];
  __shared__ float redbuf[8];

  int bi, bj;
  decode_tri((int)blockIdx.x, bi, bj);
  const int rowBase = bi * BT;
  const int colBase = bj * BT;
  const float* Ap = (rowBase < NHALF) ? src + (size_t)rowBase * DDIM
                                      : tgt + (size_t)(rowBase - NHALF) * DDIM;
  const float* Bp = (colBase < NHALF) ? src + (size_t)colBase * DDIM
                                      : tgt + (size_t)(colBase - NHALF) * DDIM;

  const int tid = threadIdx.x;
  if (tid < BT)      sqA[tid]      = sq[rowBase + tid];
  else               sqB[tid - BT] = sq[colBase + (tid - BT)];

  const int lane = tid & 31;
  const int wave = tid >> 5;
  const int wRow = (wave & 3) * 32;   // 4 wave-rows of 32
  const int wCol = (wave >> 2) * 64;  // 2 wave-cols of 64
  const int m16  = lane & 15;
  const int ko   = (lane >> 4) * 2;

  v8f acc[2][4];
#pragma unroll
  for (int rt = 0; rt < 2; ++rt)
#pragma unroll
    for (int ct = 0; ct < 4; ++ct)
#pragma unroll
      for (int e = 0; e < 8; ++e) acc[rt][ct][e] = 0.0f;

  for (int kt = 0; kt < DDIM; kt += KT) {
    __syncthreads();  // also covers initial sqA/sqB writes
    // stage 128x32 slices of A and B (float4, coalesced)
    for (int i = tid; i < BT * (KT / 4); i += 256) {
      int r  = i >> 3;       // KT/4 == 8 chunks per row
      int c4 = (i & 7) * 4;
      *(v4f*)(&lA[r * LDSS + c4]) = *(const v4f*)(Ap + (size_t)r * DDIM + kt + c4);
      *(v4f*)(&lB[r * LDSS + c4]) = *(const v4f*)(Bp + (size_t)r * DDIM + kt + c4);
    }
    __syncthreads();
#pragma unroll
    for (int ks = 0; ks < KT; ks += 4) {
      v2f afrag[2], bfrag[4];
#pragma unroll
      for (int rt = 0; rt < 2; ++rt)
        afrag[rt] = *(const v2f*)(&lA[(wRow + rt * 16 + m16) * LDSS + ks + ko]);
#pragma unroll
      for (int ct = 0; ct < 4; ++ct)
        bfrag[ct] = *(const v2f*)(&lB[(wCol + ct * 16 + m16) * LDSS + ks + ko]);
#pragma unroll
      for (int rt = 0; rt < 2; ++rt)
#pragma unroll
        for (int ct = 0; ct < 4; ++ct)
          acc[rt][ct] = __builtin_amdgcn_wmma_f32_16x16x4_f32(
              false, afrag[rt], false, bfrag[ct],
              (short)0, acc[rt][ct], false, false);
    }
  }

  // ------------------------------------------------------------- epilogue
  float s0 = 0.0f;
  if (PASS == 2) {
    // bw0 = sum(l2)/(N^2-N) / 2^(5/2=2);  exp(-l2/bw_i) = exp2(-l2*log2e/bw_i)
    s0 = 1.44269504088896340736f / bwp[0];
  }
  float local = 0.0f;
  const int mrow = (lane >> 4) * 8;
#pragma unroll
  for (int rt = 0; rt < 2; ++rt) {
#pragma unroll
    for (int ct = 0; ct < 4; ++ct) {
#pragma unroll
      for (int r = 0; r < 8; ++r) {
        int lm = wRow + rt * 16 + mrow + r;   // local row in [0,128)
        int ln = wCol + ct * 16 + m16;        // local col in [0,128)
        float g  = acc[rt][ct][r];
        float l2 = fmaxf(sqA[lm] + sqB[ln] - 2.0f * g, 0.0f);
        if (PASS == 1) {
          local += l2;
        } else {
          float s = s0;
#pragma unroll
          for (int i = 0; i < 5; ++i) { local += exp2f(-l2 * s); s *= 0.5f; }
        }
      }
    }
  }
  if (PASS == 1 && bi != bj) local *= 2.0f;  // symmetric mirror

#pragma unroll
  for (int off = 16; off > 0; off >>= 1) local += __shfl_xor(local, off, 32);
  if (lane == 0) redbuf[wave] = local;
  __syncthreads();
  if (tid == 0) {
    float s = 0.0f;
#pragma unroll
    for (int w = 0; w < 8; ++w) s += redbuf[w];
    partial[blockIdx.x] = s;
  }
}

// ----------------------------------------------- bandwidth reduction (det.)
__global__ __launch_bounds__(256)
void mmd_reduce_bw(const float* __restrict__ partial, float* __restrict__ bwout) {
  __shared__ float buf[256];
  float s = 0.0f;
  for (int i = threadIdx.x; i < NBLK; i += 256) s += partial[i];
  buf[threadIdx.x] = s;
  __syncthreads();
  for (int st = 128; st > 0; st >>= 1) {
    if ((int)threadIdx.x < st) buf[threadIdx.x] += buf[threadIdx.x + st];
    __syncthreads();
  }
  if (threadIdx.x == 0) {
    double total = (double)buf[0];
    const double Ns = (double)NTOT;
    double bw = total / (Ns * Ns - Ns);
    bw = bw / 4.0;  // KERNEL_MUL ^ (KERNEL_NUM // 2) = 2^2
    bwout[0] = (float)bw;
  }
}

// -------------------------------------------------- final quadrant combine
__global__ __launch_bounds__(256)
void mmd_final(const float* __restrict__ partial2, float* __restrict__ out) {
  __shared__ float q[4][256];
  float ql[4] = {0.0f, 0.0f, 0.0f, 0.0f};
  const int HB = NHALF / BT;  // 32: block index boundary between x and y
  for (int t = threadIdx.x; t < NBLK; t += 256) {
    int bi, bj;
    decode_tri(t, bi, bj);
    float T = partial2[t];
    int qa = ((bi >= HB) << 1) | (bj >= HB);
    ql[qa] += T;
    if (bi != bj) {
      int qb = ((bj >= HB) << 1) | (bi >= HB);
      ql[qb] += T;
    }
  }
#pragma unroll
  for (int c = 0; c < 4; ++c) q[c][threadIdx.x] = ql[c];
  __syncthreads();
  for (int st = 128; st > 0; st >>= 1) {
    if ((int)threadIdx.x < st)
#pragma unroll
      for (int c = 0; c < 4; ++c) q[c][threadIdx.x] += q[c][threadIdx.x + st];
    __syncthreads();
  }
  if (threadIdx.x == 0) {
    float xx = q[0][0], xy = q[1][0], yx = q[2][0], yy = q[3][0];
    const float inv = 1.0f / ((float)NHALF * (float)NHALF);  // n == m
    out[0] = (xx + yy - xy - yx) * inv;
  }
}

// ---------------------------------------------------------------- launcher
extern "C" void kernel_launch(void* const* d_in, const int* in_sizes, int n_in,
                              void* d_out, int out_size, void* d_ws, size_t ws_size,
                              hipStream_t stream) {
  (void)in_sizes; (void)n_in; (void)out_size; (void)ws_size;
  const float* src = (const float*)d_in[0];
  const float* tgt = (const float*)d_in[1];
  float* out = (float*)d_out;
  float* wsf = (float*)d_ws;         // needs >= 64 KB workspace
  float* bw  = wsf + WS_BW;
  float* p1  = wsf + WS_P1;
  float* p2  = wsf + WS_P2;
  float* sq  = wsf + WS_SQ;

  mmd_sq<<<NTOT / 8, 256, 0, stream>>>(src, tgt, sq);
  mmd_gemm<1><<<NBLK, 256, 0, stream>>>(src, tgt, sq, p1, bw);
  mmd_reduce_bw<<<1, 256, 0, stream>>>(p1, bw);
  mmd_gemm<2><<<NBLK, 256, 0, stream>>>(src, tgt, sq, p2, bw);
  mmd_final<<<1, 256, 0, stream>>>(p2, out);
}